// DistributedSpectralConvNd_85873576116714
// MI455X (gfx1250) — compile-verified
//
#include <hip/hip_runtime.h>

// MI455X / gfx1250, wave32. Matrix math via V_WMMA_F32_16X16X4_F32.
// DFT-as-GEMM pipeline; twiddle tables stored transposed ([col][K]) so each
// B-fragment is one 8-byte load, with pre-negated sin tables (f32 WMMA has no
// A/B NEG bits per ISA 7.12).

typedef __attribute__((ext_vector_type(2))) float v2f;
typedef __attribute__((ext_vector_type(8))) float v8f;

#define WMMA4(a, b, c) \
  __builtin_amdgcn_wmma_f32_16x16x4_f32(false, (a), false, (b), (short)0, (c), false, false)

// ---------------------------------------------------------------------------
// Table layout (floats from tab base). All transposed: [col][K].
// ---------------------------------------------------------------------------
#define TZC   0        // [16][64] cos(2pi j n/64)          (z fwd, col=kz, K=z)
#define TZNS  1024     // [16][64] -sin
#define TFC   2048     // [32][64] cos(2pi j f/64)          (x/y fwd, col=t, K=pos)
#define TFS   4096     // [32][64] +sin
#define TFNS  6144     // [32][64] -sin
#define TIC   8192     // [64][32] cos(2pi f n/64)/64       (x/y inv, col=n, K=mode)
#define TIS   10240    // [64][32] +sin/64
#define TINS  12288    // [64][32] -sin/64
#define TZIT  14336    // [64][32] irfft weights, col=z pos, K: 0..15 re, 16..31 im
#define WS_TAB 16777216u

// ---------------------------------------------------------------------------
// Table builder
// ---------------------------------------------------------------------------
__global__ void k_tables(float* __restrict__ tab) {
  const int t = threadIdx.x;
  const float PI2 = 6.283185307179586f;
  const float inv64 = 1.0f / 64.0f;
  for (int i = t; i < 1024; i += 256) {           // z forward
    const int n = i >> 6, j = i & 63;
    const float th = PI2 * (float)((n * j) & 63) * inv64;
    tab[TZC + i]  = __cosf(th);
    tab[TZNS + i] = -__sinf(th);
  }
  for (int i = t; i < 2048; i += 256) {           // x/y forward
    const int c = i >> 6, j = i & 63;
    const int f = (c < 16) ? c : (c + 32);
    const float th = PI2 * (float)((f * j) & 63) * inv64;
    const float sv = __sinf(th);
    tab[TFC + i]  = __cosf(th);
    tab[TFS + i]  = sv;
    tab[TFNS + i] = -sv;
  }
  for (int i = t; i < 2048; i += 256) {           // x/y inverse (scaled 1/64)
    const int n = i >> 5, u = i & 31;
    const int f = (u < 16) ? u : (u + 32);
    const float th = PI2 * (float)((f * n) & 63) * inv64;
    const float sv = __sinf(th) * inv64;
    tab[TIC + i]  = __cosf(th) * inv64;
    tab[TIS + i]  = sv;
    tab[TINS + i] = -sv;
  }
  for (int i = t; i < 2048; i += 256) {           // z inverse (c2r)
    const int n = i >> 5, k = i & 31;
    float v;
    if (k < 16) {
      v = (k == 0) ? inv64
                   : 2.0f * __cosf(PI2 * (float)((k * n) & 63) * inv64) * inv64;
    } else {
      const int kk = k - 16;
      v = (kk == 0) ? 0.0f
                    : -2.0f * __sinf(PI2 * (float)((kk * n) & 63) * inv64) * inv64;
    }
    tab[TZIT + i] = v;
  }
}

// ---------------------------------------------------------------------------
// Fused complex-forward accumulator chains (K=64, 32 cols = 2 n-tiles).
// e^{-i th}: Or = Ar*C + Ai*S ; Oi = Ai*C + Ar*(-S)
// ---------------------------------------------------------------------------
__device__ __forceinline__ void fwd32(const float* __restrict__ Ar,
                                      const float* __restrict__ Ai,
                                      const float* __restrict__ tab,
                                      v8f& oR0, v8f& oI0, v8f& oR1, v8f& oI1) {
  const int l = threadIdx.x & 31;
  const int m = l & 15, kh = (l >> 4) * 2;
  const float* c0 = tab + TFC + m * 64;
  const float* s0 = tab + TFS + m * 64;
  const float* n0 = tab + TFNS + m * 64;
  const float* c1 = c0 + 1024;
  const float* s1 = s0 + 1024;
  const float* n1 = n0 + 1024;
#pragma unroll
  for (int k = 0; k < 64; k += 4) {
    const int kb = k + kh;
    const v2f ar = *(const v2f*)&Ar[m * 64 + kb];
    const v2f ai = *(const v2f*)&Ai[m * 64 + kb];
    const v2f bc0 = *(const v2f*)&c0[kb];
    const v2f bs0 = *(const v2f*)&s0[kb];
    const v2f bn0 = *(const v2f*)&n0[kb];
    const v2f bc1 = *(const v2f*)&c1[kb];
    const v2f bs1 = *(const v2f*)&s1[kb];
    const v2f bn1 = *(const v2f*)&n1[kb];
    oR0 = WMMA4(ar, bc0, oR0); oR0 = WMMA4(ai, bs0, oR0);
    oI0 = WMMA4(ai, bc0, oI0); oI0 = WMMA4(ar, bn0, oI0);
    oR1 = WMMA4(ar, bc1, oR1); oR1 = WMMA4(ai, bs1, oR1);
    oI1 = WMMA4(ai, bc1, oI1); oI1 = WMMA4(ar, bn1, oI1);
  }
}

// ---------------------------------------------------------------------------
// Fused complex-inverse accumulator chains (K=32, 64 cols = 4 n-tiles).
// e^{+i th}/64: Or = Ar*c + Ai*(-s) ; Oi = Ar*s + Ai*c
// ---------------------------------------------------------------------------
__device__ __forceinline__ void inv64(const float* __restrict__ Ar,
                                      const float* __restrict__ Ai,
                                      const float* __restrict__ tab,
                                      v8f (&oR)[4], v8f (&oI)[4]) {
  const int l = threadIdx.x & 31;
  const int m = l & 15, kh = (l >> 4) * 2;
#pragma unroll
  for (int k = 0; k < 32; k += 4) {
    const int kb = k + kh;
    const v2f ar = *(const v2f*)&Ar[m * 32 + kb];
    const v2f ai = *(const v2f*)&Ai[m * 32 + kb];
#pragma unroll
    for (int nt = 0; nt < 4; ++nt) {
      const int col = nt * 16 + m;
      const v2f c  = *(const v2f*)&tab[TIC + col * 32 + kb];
      const v2f s  = *(const v2f*)&tab[TIS + col * 32 + kb];
      const v2f ns = *(const v2f*)&tab[TINS + col * 32 + kb];
      oR[nt] = WMMA4(ar, c, oR[nt]);  oR[nt] = WMMA4(ai, ns, oR[nt]);
      oI[nt] = WMMA4(ar, s, oI[nt]);  oI[nt] = WMMA4(ai, c, oI[nt]);
    }
  }
}

// ---------------------------------------------------------------------------
// Stage 1: forward real DFT along z. rows r = ((b*CI+ci)*64+x)*64+y, K=64.
// out[(base*16+kz)*64+y], base=(b,ci,x).
// ---------------------------------------------------------------------------
__global__ void k_zfwd(const float* __restrict__ x,
                       float* __restrict__ s1re, float* __restrict__ s1im,
                       const float* __restrict__ tab) {
  __shared__ __align__(16) float As[16 * 64];
  const int r0 = blockIdx.x * 16;
  const int l = threadIdx.x;
  for (int i = l; i < 1024; i += 32) As[i] = x[r0 * 64 + i];
  __builtin_prefetch(&x[(r0 + 16) * 64], 0, 1);
  __syncthreads();
  const int m = l & 15, kh = (l >> 4) * 2;
  const float* bC = tab + TZC + m * 64;
  const float* bN = tab + TZNS + m * 64;
  v8f aRe = {}, aIm = {};
#pragma unroll
  for (int k = 0; k < 64; k += 4) {
    const int kb = k + kh;
    const v2f a = *(const v2f*)&As[m * 64 + kb];
    const v2f c = *(const v2f*)&bC[kb];
    const v2f s = *(const v2f*)&bN[kb];
    aRe = WMMA4(a, c, aRe);
    aIm = WMMA4(a, s, aIm);
  }
  const int n = l & 15, m0 = (l >> 4) * 8;
  for (int v = 0; v < 8; ++v) {
    const int r = r0 + m0 + v;
    const int base = r >> 6, y = r & 63;
    const int idx = (base * 16 + n) * 64 + y;
    s1re[idx] = aRe[v];
    s1im[idx] = aIm[v];
  }
}

// ---------------------------------------------------------------------------
// Stage 2: forward complex DFT along y. rows r = (bc*64+x)*16+kz, K=64.
// out[((bc*32+t)*16+kz)*64+x]
// ---------------------------------------------------------------------------
__global__ void k_yfwd(const float* __restrict__ s1re, const float* __restrict__ s1im,
                       float* __restrict__ s2re, float* __restrict__ s2im,
                       const float* __restrict__ tab) {
  __shared__ __align__(16) float Ar[16 * 64], Ai[16 * 64];
  const int r0 = blockIdx.x * 16;
  const int l = threadIdx.x;
  for (int i = l; i < 1024; i += 32) {
    Ar[i] = s1re[r0 * 64 + i];
    Ai[i] = s1im[r0 * 64 + i];
  }
  __syncthreads();
  v8f oR0 = {}, oI0 = {}, oR1 = {}, oI1 = {};
  fwd32(Ar, Ai, tab, oR0, oI0, oR1, oI1);
  const int n = l & 15, m0 = (l >> 4) * 8;
  for (int v = 0; v < 8; ++v) {
    const int r = r0 + m0 + v;
    const int bc = r >> 10, xx = (r >> 4) & 63, kz = r & 15;
    const int i0 = ((bc * 32 + n) * 16 + kz) * 64 + xx;
    const int i1 = ((bc * 32 + n + 16) * 16 + kz) * 64 + xx;
    s2re[i0] = oR0[v]; s2im[i0] = oI0[v];
    s2re[i1] = oR1[v]; s2im[i1] = oI1[v];
  }
}

// ---------------------------------------------------------------------------
// Stage 3: forward complex DFT along x. rows r = (bc*32+t)*16+kz, K=64.
// out[(((u*32+t)*16+kz)*4+b)*32+ci]  (kz,b,ci contiguous per (u,t) slab)
// ---------------------------------------------------------------------------
__global__ void k_xfwd(const float* __restrict__ s2re, const float* __restrict__ s2im,
                       float* __restrict__ xhr, float* __restrict__ xhi,
                       const float* __restrict__ tab) {
  __shared__ __align__(16) float Ar[16 * 64], Ai[16 * 64];
  const int r0 = blockIdx.x * 16;
  const int l = threadIdx.x;
  for (int i = l; i < 1024; i += 32) {
    Ar[i] = s2re[r0 * 64 + i];
    Ai[i] = s2im[r0 * 64 + i];
  }
  __syncthreads();
  v8f oR0 = {}, oI0 = {}, oR1 = {}, oI1 = {};
  fwd32(Ar, Ai, tab, oR0, oI0, oR1, oI1);
  const int n = l & 15, m0 = (l >> 4) * 8;
  for (int v = 0; v < 8; ++v) {
    const int r = r0 + m0 + v;
    const int bc = r >> 9, t = (r >> 4) & 31, kz = r & 15;
    const int b = bc >> 5, ci = bc & 31;
    const int i0 = (((n * 32 + t) * 16 + kz) * 4 + b) * 32 + ci;
    const int i1 = ((((n + 16) * 32 + t) * 16 + kz) * 4 + b) * 32 + ci;
    xhr[i0] = oR0[v]; xhi[i0] = oI0[v];
    xhr[i1] = oR1[v]; xhi[i1] = oI1[v];
  }
}

// ---------------------------------------------------------------------------
// Stage 4: spectral multiply. One block per (u,t); Xhat slab (2048 fl/plane,
// contiguous) staged in LDS and reused by all co; weight reads coalesced in kz.
// out[(((t*16+kz)*4+b)*32+co)*32 + u]  (kx contiguous for stage 5)
// ---------------------------------------------------------------------------
__global__ void k_spec(const float* __restrict__ xhr, const float* __restrict__ xhi,
                       const float* __restrict__ w0, const float* __restrict__ w1,
                       const float* __restrict__ w2, const float* __restrict__ w3,
                       float* __restrict__ yhr, float* __restrict__ yhi) {
  __shared__ __align__(16) float Lr[2048], Li[2048];   // [kz][b][ci]
  const int u = blockIdx.x >> 5;
  const int t = blockIdx.x & 31;
  const int tid = threadIdx.x;                         // 256 threads
  const int slab = (u * 32 + t) * 2048;
  for (int i = tid; i < 2048; i += 256) {
    Lr[i] = xhr[slab + i];
    Li[i] = xhi[slab + i];
  }
  __syncthreads();
  const float* __restrict__ w = (u < 16) ? ((t < 16) ? w0 : w1)
                                         : ((t < 16) ? w2 : w3);
  const int a = u & 15, bb = t & 15;
  const int kz = tid & 15, coh = tid >> 4;             // coh: 0..15
  for (int ch = 0; ch < 2; ++ch) {
    const int co = coh + ch * 16;
    float orr[4] = {0.f, 0.f, 0.f, 0.f};
    float oii[4] = {0.f, 0.f, 0.f, 0.f};
    for (int ci = 0; ci < 32; ++ci) {
      const float* wp =
          w + (size_t)((((ci * 32 + co) * 16 + a) * 16 + bb) * 16 + kz) * 2;
      const float wr = wp[0], wi = wp[1];
#pragma unroll
      for (int b = 0; b < 4; ++b) {
        const float ar = Lr[(kz * 4 + b) * 32 + ci];
        const float ai = Li[(kz * 4 + b) * 32 + ci];
        orr[b] += ar * wr - ai * wi;
        oii[b] += ar * wi + ai * wr;
      }
    }
#pragma unroll
    for (int b = 0; b < 4; ++b) {
      const int oidx = (((t * 16 + kz) * 4 + b) * 32 + co) * 32 + u;
      yhr[oidx] = orr[b];
      yhi[oidx] = oii[b];
    }
  }
}

// ---------------------------------------------------------------------------
// Stage 5: inverse complex DFT along x. rows r = ((t*16+kz)*4+b)*32+co, K=32.
// out[(((b*32+co)*64+x)*16+kz)*32 + t]
// ---------------------------------------------------------------------------
__global__ void k_xinv(const float* __restrict__ yhr, const float* __restrict__ yhi,
                       float* __restrict__ s5re, float* __restrict__ s5im,
                       const float* __restrict__ tab) {
  __shared__ __align__(16) float Ar[16 * 32], Ai[16 * 32];
  const int r0 = blockIdx.x * 16;
  const int l = threadIdx.x;
  for (int i = l; i < 512; i += 32) {
    Ar[i] = yhr[r0 * 32 + i];
    Ai[i] = yhi[r0 * 32 + i];
  }
  __syncthreads();
  v8f oR[4] = {}, oI[4] = {};
  inv64(Ar, Ai, tab, oR, oI);
  const int n = l & 15, m0 = (l >> 4) * 8;
  for (int v = 0; v < 8; ++v) {
    const int r = r0 + m0 + v;
    const int t = r >> 11, kz = (r >> 7) & 15, b = (r >> 5) & 3, co = r & 31;
    for (int nt = 0; nt < 4; ++nt) {
      const int xx = nt * 16 + n;
      const int idx = (((b * 32 + co) * 64 + xx) * 16 + kz) * 32 + t;
      s5re[idx] = oR[nt][v];
      s5im[idx] = oI[nt][v];
    }
  }
}

// ---------------------------------------------------------------------------
// Stage 6: inverse complex DFT along y. rows r = ((b*32+co)*64+x)*16+kz, K=32.
// out[(bcx*64+y)*16 + kz]  (kz contiguous for stage 7)
// ---------------------------------------------------------------------------
__global__ void k_yinv(const float* __restrict__ s5re, const float* __restrict__ s5im,
                       float* __restrict__ s6re, float* __restrict__ s6im,
                       const float* __restrict__ tab) {
  __shared__ __align__(16) float Ar[16 * 32], Ai[16 * 32];
  const int r0 = blockIdx.x * 16;
  const int l = threadIdx.x;
  for (int i = l; i < 512; i += 32) {
    Ar[i] = s5re[r0 * 32 + i];
    Ai[i] = s5im[r0 * 32 + i];
  }
  __syncthreads();
  v8f oR[4] = {}, oI[4] = {};
  inv64(Ar, Ai, tab, oR, oI);
  const int n = l & 15, m0 = (l >> 4) * 8;
  for (int v = 0; v < 8; ++v) {
    const int r = r0 + m0 + v;
    const int bcx = r >> 4, kz = r & 15;
    for (int nt = 0; nt < 4; ++nt) {
      const int y = nt * 16 + n;
      const int idx = (bcx * 64 + y) * 16 + kz;
      s6re[idx] = oR[nt][v];
      s6im[idx] = oI[nt][v];
    }
  }
}

// ---------------------------------------------------------------------------
// Stage 7: inverse real DFT along z (c2r). rows r = ((b*32+co)*64+x)*64+y.
// A row = [re(kz 0..15) | im(kz 0..15)], K=32; B = TZIT[64][32].
// ---------------------------------------------------------------------------
__global__ void k_zinv(const float* __restrict__ s6re, const float* __restrict__ s6im,
                       float* __restrict__ out, const float* __restrict__ tab) {
  __shared__ __align__(16) float As[16 * 32];
  const int r0 = blockIdx.x * 16;
  const int l = threadIdx.x;
  for (int i = l; i < 512; i += 32) {
    const int row = i >> 5, c = i & 31;
    As[i] = (c < 16) ? s6re[(r0 + row) * 16 + c]
                     : s6im[(r0 + row) * 16 + (c - 16)];
  }
  __syncthreads();
  const int m = l & 15, kh = (l >> 4) * 2;
  const int n = l & 15, m0 = (l >> 4) * 8;
#pragma unroll
  for (int nt = 0; nt < 4; ++nt) {
    v8f acc = {};
#pragma unroll
    for (int k = 0; k < 32; k += 4) {
      const int kb = k + kh;
      const v2f a = *(const v2f*)&As[m * 32 + kb];
      const v2f b = *(const v2f*)&tab[TZIT + (nt * 16 + m) * 32 + kb];
      acc = WMMA4(a, b, acc);
    }
    for (int v = 0; v < 8; ++v) {
      const int r = r0 + m0 + v;
      out[r * 64 + nt * 16 + n] = acc[v];
    }
  }
}

// ---------------------------------------------------------------------------
// Launch
// ---------------------------------------------------------------------------
extern "C" void kernel_launch(void* const* d_in, const int* in_sizes, int n_in,
                              void* d_out, int out_size, void* d_ws, size_t ws_size,
                              hipStream_t stream) {
  const float* x  = (const float*)d_in[0];
  const float* w0 = (const float*)d_in[1];
  const float* w1 = (const float*)d_in[2];
  const float* w2 = (const float*)d_in[3];
  const float* w3 = (const float*)d_in[4];
  float* out = (float*)d_out;                 // scratch until stage 7
  float* ws  = (float*)d_ws;
  float* tab = ws + WS_TAB;

  k_tables<<<1, 256, 0, stream>>>(tab);

  // stage 1: x -> S1 planes in d_out[0 / 8388608]
  k_zfwd<<<32768, 32, 0, stream>>>(x, out, out + 8388608, tab);
  // stage 2: S1 -> S2 planes in ws[0 / 4194304]
  k_yfwd<<<8192, 32, 0, stream>>>(out, out + 8388608, ws, ws + 4194304, tab);
  // stage 3: S2 -> Xhat planes in d_out[0 / 2097152]
  k_xfwd<<<4096, 32, 0, stream>>>(ws, ws + 4194304, out, out + 2097152, tab);
  // stage 4: Xhat * W -> Yhat planes in ws[0 / 2097152]
  k_spec<<<1024, 256, 0, stream>>>(out, out + 2097152, w0, w1, w2, w3,
                                   ws, ws + 2097152);
  // stage 5: Yhat -> S5 planes in d_out[0 / 4194304]
  k_xinv<<<4096, 32, 0, stream>>>(ws, ws + 2097152, out, out + 4194304, tab);
  // stage 6: S5 -> S6 planes in ws[0 / 8388608]
  k_yinv<<<8192, 32, 0, stream>>>(out, out + 4194304, ws, ws + 8388608, tab);
  // stage 7: S6 -> final real output (fully overwrites d_out)
  k_zinv<<<32768, 32, 0, stream>>>(ws, ws + 8388608, out, tab);
}